// GNNModel_11613591569246
// MI455X (gfx1250) — compile-verified
//
#include <hip/hip_runtime.h>

typedef float v2f __attribute__((ext_vector_type(2)));
typedef float v8f __attribute__((ext_vector_type(8)));

#define D_FEAT 128   // D_IN == D_HID == 128

// ---------------------------------------------------------------------------
// Degree / normalization kernels
// ---------------------------------------------------------------------------
__global__ void deg_init_kernel(float* __restrict__ deg, int n) {
    int i = blockIdx.x * blockDim.x + threadIdx.x;
    if (i < n) deg[i] = 1.0f;   // self-loop contributes 1
}

__global__ void deg_count_kernel(const long long* __restrict__ dst,
                                 float* __restrict__ deg, int nE) {
    int i = blockIdx.x * blockDim.x + threadIdx.x;
    if (i < nE) unsafeAtomicAdd(&deg[(int)dst[i]], 1.0f);
}

__global__ void deg_to_inv_kernel(float* __restrict__ deg, int n) {
    int i = blockIdx.x * blockDim.x + threadIdx.x;
    if (i < n) deg[i] = rsqrtf(deg[i]);   // deg >= 1 always (self loop)
}

// ---------------------------------------------------------------------------
// GEMM:  Hs[row, :] = (X[row, :] @ W) * inv[row]   (also duplicated into S,
// which seeds the aggregation with the self-loop term).
// One wave computes one 16-row M-tile across the full DOUT width using
// V_WMMA_F32_16X16X4_F32 (fp32 in, fp32 out -> matches reference numerics).
//
// f32 16x16x4 operand layout (ISA 7.12.2):
//   A (16x4, 2 VGPR): lanes 0-15 rows M=lane, K={0,1}; lanes 16-31 K={2,3}
//   B (4x16, 2 VGPR): VGPR0 rows K=0 (lanes 0-15) / K=2 (lanes 16-31),
//                     VGPR1 rows K=1 / K=3, N = lane&15
//   C/D (16x16, 8 VGPR): VGPR v -> row v (lanes 0-15), row v+8 (lanes 16-31)
// ---------------------------------------------------------------------------
template <int DOUT>
__launch_bounds__(256)
__global__ void gemm_scale_kernel(const float* __restrict__ X,
                                  const float* __restrict__ W,
                                  const float* __restrict__ inv,
                                  float* __restrict__ Hs,
                                  float* __restrict__ S,
                                  int mtiles) {
    __shared__ float Ws[D_FEAT * DOUT];   // 64 KB (DOUT=128) / 32 KB (DOUT=64)
    for (int i = threadIdx.x; i < D_FEAT * DOUT; i += 256) Ws[i] = W[i];
    __syncthreads();

    const int lane = threadIdx.x & 31;
    const int wave = threadIdx.x >> 5;
    const int mt   = blockIdx.x * 8 + wave;
    if (mt >= mtiles) return;             // wave-uniform: EXEC stays all-ones

    const int r  = lane & 15;             // row-in-tile / column index
    const int hf = lane >> 4;             // lane half selects K pairs

    const float* __restrict__ arow = X + (size_t)(mt * 16 + r) * D_FEAT;

    constexpr int NT = DOUT / 16;
    v8f acc[NT] = {};

    for (int k = 0; k < D_FEAT; k += 4) {
        v2f a;
        a.x = arow[k + 2 * hf];
        a.y = arow[k + 2 * hf + 1];
#pragma unroll
        for (int t = 0; t < NT; ++t) {
            v2f b;
            b.x = Ws[(k + 2 * hf) * DOUT + t * 16 + r];
            b.y = Ws[(k + 2 * hf + 1) * DOUT + t * 16 + r];
            acc[t] = __builtin_amdgcn_wmma_f32_16x16x4_f32(
                false, a, false, b, (short)0, acc[t], false, false);
        }
    }

#pragma unroll
    for (int v = 0; v < 8; ++v) {
        const int row = mt * 16 + v + 8 * hf;
        const float sc = inv[row];
#pragma unroll
        for (int t = 0; t < NT; ++t) {
            const float val = acc[t][v] * sc;
            const size_t off = (size_t)row * DOUT + t * 16 + r;
            Hs[off] = val;   // message buffer, read by scatter
            S[off]  = val;   // accumulator seeded with self-loop term
        }
    }
}

// ---------------------------------------------------------------------------
// Edge scatter-add: S[dst, c] += Hs[src, c]  (pre-scaled messages)
// ---------------------------------------------------------------------------
template <int DOUT>
__global__ void scatter_kernel(const float* __restrict__ Hs,
                               float* __restrict__ S,
                               const long long* __restrict__ src,
                               const long long* __restrict__ dst,
                               int nE) {
    constexpr int SH = (DOUT == 128) ? 7 : 6;
    const long long idx = (long long)blockIdx.x * blockDim.x + threadIdx.x;
    const int e = (int)(idx >> SH);
    const int c = (int)(idx & (DOUT - 1));
    if (e >= nE) return;
    const int s = (int)src[e];
    const int d = (int)dst[e];
    unsafeAtomicAdd(&S[(size_t)d * DOUT + c], Hs[(size_t)s * DOUT + c]);
}

// ---------------------------------------------------------------------------
// Epilogue: out = [relu](S * inv[row] + bias[c]) [+ residual]
// ---------------------------------------------------------------------------
template <int DOUT, bool RELU, bool RES>
__global__ void finalize_kernel(const float* __restrict__ S,
                                const float* __restrict__ inv,
                                const float* __restrict__ bias,
                                const float* __restrict__ resid,
                                float* __restrict__ out, int n) {
    constexpr int SH = (DOUT == 128) ? 7 : 6;
    const long long i = (long long)blockIdx.x * blockDim.x + threadIdx.x;
    if (i >= (long long)n * DOUT) return;
    const int row = (int)(i >> SH);
    const int c   = (int)(i & (DOUT - 1));
    float v = S[i] * inv[row] + bias[c];
    if (RELU) v = fmaxf(v, 0.0f);
    if (RES)  v += resid[i];
    out[i] = v;
}

// ---------------------------------------------------------------------------
// Host orchestration
// ---------------------------------------------------------------------------
static inline unsigned cdiv(long long a, long long b) { return (unsigned)((a + b - 1) / b); }

extern "C" void kernel_launch(void* const* d_in, const int* in_sizes, int n_in,
                              void* d_out, int out_size, void* d_ws, size_t ws_size,
                              hipStream_t stream) {
    const float*     x    = (const float*)d_in[0];
    const long long* ei   = (const long long*)d_in[1];   // int64 (2, E)
    const float*     W0   = (const float*)d_in[2];
    const float*     b0   = (const float*)d_in[3];
    const float*     Wres = (const float*)d_in[4];       // (2, 128, 128)
    const float*     bres = (const float*)d_in[5];       // (2, 128)
    const float*     Wf   = (const float*)d_in[6];       // (128, 64)
    const float*     bf   = (const float*)d_in[7];
    float*           out  = (float*)d_out;

    const int N = in_sizes[0] / D_FEAT;   // 100000
    const int E = in_sizes[1] / 2;        // 1600000
    const long long* srcI = ei;
    const long long* dstI = ei + E;

    const size_t NB = (size_t)N * D_FEAT;
    float* B0  = (float*)d_ws;        // rotating feature buffers (N x 128)
    float* B1  = B0 + NB;
    float* B2  = B1 + NB;
    float* inv = B2 + NB;             // N floats

    const dim3 blk(256);
    const int mtiles = N / 16;                         // 6250 (N % 16 == 0)
    const unsigned gGemm = cdiv(mtiles, 8);
    const unsigned gN    = cdiv(N, 256);
    const unsigned gE    = cdiv(E, 256);
    const unsigned gS128 = cdiv((long long)E * 128, 256);
    const unsigned gS64  = cdiv((long long)E * 64, 256);
    const unsigned gF128 = cdiv((long long)N * 128, 256);
    const unsigned gF64  = cdiv((long long)N * 64, 256);

    // ---- normalization: inv = rsqrt(1 + in_degree) ----
    deg_init_kernel<<<gN, blk, 0, stream>>>(inv, N);
    deg_count_kernel<<<gE, blk, 0, stream>>>(dstI, inv, E);
    deg_to_inv_kernel<<<gN, blk, 0, stream>>>(inv, N);

    // ---- layer 0: relu(conv(x, W0) + b0) -> B1 ----
    gemm_scale_kernel<128><<<gGemm, blk, 0, stream>>>(x, W0, inv, B0, B1, mtiles);
    scatter_kernel<128><<<gS128, blk, 0, stream>>>(B0, B1, srcI, dstI, E);
    finalize_kernel<128, true, false><<<gF128, blk, 0, stream>>>(B1, inv, b0, nullptr, B1, N);

    // ---- residual block 0: relu(conv(B1)) + B1 -> B2 ----
    gemm_scale_kernel<128><<<gGemm, blk, 0, stream>>>(B1, Wres, inv, B0, B2, mtiles);
    scatter_kernel<128><<<gS128, blk, 0, stream>>>(B0, B2, srcI, dstI, E);
    finalize_kernel<128, true, true><<<gF128, blk, 0, stream>>>(B2, inv, bres, B1, B2, N);

    // ---- residual block 1: relu(conv(B2)) + B2 -> B1 ----
    gemm_scale_kernel<128><<<gGemm, blk, 0, stream>>>(B2, Wres + (size_t)128 * 128, inv, B0, B1, mtiles);
    scatter_kernel<128><<<gS128, blk, 0, stream>>>(B0, B1, srcI, dstI, E);
    finalize_kernel<128, true, true><<<gF128, blk, 0, stream>>>(B1, inv, bres + 128, B2, B1, N);

    // ---- final conv (128 -> 64), no activation -> d_out ----
    gemm_scale_kernel<64><<<gGemm, blk, 0, stream>>>(B1, Wf, inv, B0, B2, mtiles);
    scatter_kernel<64><<<gS64, blk, 0, stream>>>(B0, B2, srcI, dstI, E);
    finalize_kernel<64, false, false><<<gF64, blk, 0, stream>>>(B2, inv, bf, nullptr, out, N);
}